// WarpingLayer_Flow_Exp_27848567947671
// MI455X (gfx1250) — compile-verified
//
#include <hip/hip_runtime.h>

typedef __attribute__((ext_vector_type(2))) float v2f;
typedef __attribute__((ext_vector_type(8))) float v8f;

#define B_  4
#define C_  128
#define H_  128
#define W_  416
#define HW_ (H_ * W_)
#define HH_ 64
#define WH_ 208
#define HWH_ (HH_ * WH_)

// Workspace layout (float units)
#define RY_OFF   0            // [128][64]            = 8192
#define RXT_OFF  8192         // [208][416]           = 86528
#define TMP_OFF  94720        // [8][128][208]        = 212992
#define FULL_OFF 307712       // [8][128][416]        = 425984
#define WS_FLOATS 733696

// ---------------------------------------------------------------------------
// Build align-corners interpolation matrices Ry (128x64) and RxT (208x416).
// Region [0, RXT_OFF+86528) must be zeroed beforehand (hipMemsetAsync).
// ---------------------------------------------------------------------------
__global__ void build_interp(float* ws) {
  int i = blockIdx.x * blockDim.x + threadIdx.x;
  if (i < H_) {
    // row i of Ry: ys = linspace(0, 63, 128)[i]
    float ys = (float)i * (63.0f / 127.0f);
    int   y0 = (int)floorf(ys);
    float wy = ys - (float)y0;
    int   y1 = y0 + 1; if (y1 > HH_ - 1) y1 = HH_ - 1;
    float* row = ws + RY_OFF + i * HH_;
    row[y0] = 1.0f - wy;
    row[y1] += wy;
  } else if (i < H_ + W_) {
    int xo = i - H_;
    float xs = (float)xo * (207.0f / 415.0f);
    int   x0 = (int)floorf(xs);
    float wx = xs - (float)x0;
    int   x1 = x0 + 1; if (x1 > WH_ - 1) x1 = WH_ - 1;
    float* RxT = ws + RXT_OFF;
    RxT[x0 * W_ + xo] = 1.0f - wx;
    RxT[x1 * W_ + xo] += wx;
  }
}

// ---------------------------------------------------------------------------
// GEMM1: tmp[tb] (128x208) = Ry (128x64) * D[tb] (64x208)
// tb in [0,8): tb&3 = batch, tb&4 = tensor select (0=disp, 1=exp)
// One wave per 16x16 tile; V_WMMA_F32_16X16X4_F32, 16 K-steps.
// ---------------------------------------------------------------------------
__global__ void resize_gemm1(const float* __restrict__ disp,
                             const float* __restrict__ expm,
                             float* __restrict__ ws) {
  const float* Ry = ws + RY_OFF;
  int tile = blockIdx.x;                       // 0..103  (8 x 13 tiles)
  int tb   = blockIdx.y;                       // 0..7
  int tm = tile / 13, tn = tile % 13;
  const float* D   = ((tb & 4) ? expm : disp) + (tb & 3) * HWH_;
  float*       Out = ws + TMP_OFF + tb * (H_ * WH_);

  int lane  = threadIdx.x;                     // wave32
  int m     = lane & 15;
  int khalf = (lane < 16) ? 0 : 2;             // A/B K placement per ISA layout
  int n     = lane & 15;

  v8f acc = {};
  for (int kk = 0; kk < 16; ++kk) {
    int k = kk * 4 + khalf;
    v2f a, b;
    a[0] = Ry[(tm * 16 + m) * HH_ + k];
    a[1] = Ry[(tm * 16 + m) * HH_ + k + 1];
    b[0] = D[(k)     * WH_ + tn * 16 + n];
    b[1] = D[(k + 1) * WH_ + tn * 16 + n];
    acc = __builtin_amdgcn_wmma_f32_16x16x4_f32(false, a, false, b,
                                                (short)0, acc, false, false);
  }
  for (int r = 0; r < 8; ++r) {
    int row = tm * 16 + ((lane < 16) ? r : (r + 8));
    Out[row * WH_ + tn * 16 + n] = acc[r];
  }
}

// ---------------------------------------------------------------------------
// GEMM2: full[tb] (128x416) = tmp[tb] (128x208) * RxT (208x416), 52 K-steps.
// ---------------------------------------------------------------------------
__global__ void resize_gemm2(float* __restrict__ ws) {
  const float* RxT = ws + RXT_OFF;
  int tile = blockIdx.x;                       // 0..207  (8 x 26 tiles)
  int tb   = blockIdx.y;                       // 0..7
  int tm = tile / 26, tn = tile % 26;
  const float* A   = ws + TMP_OFF  + tb * (H_ * WH_);
  float*       Out = ws + FULL_OFF + tb * HW_;

  int lane  = threadIdx.x;
  int m     = lane & 15;
  int khalf = (lane < 16) ? 0 : 2;
  int n     = lane & 15;

  v8f acc = {};
  for (int kk = 0; kk < 52; ++kk) {
    int k = kk * 4 + khalf;
    v2f a, b;
    a[0] = A[(tm * 16 + m) * WH_ + k];
    a[1] = A[(tm * 16 + m) * WH_ + k + 1];
    b[0] = RxT[(k)     * W_ + tn * 16 + n];
    b[1] = RxT[(k + 1) * W_ + tn * 16 + n];
    acc = __builtin_amdgcn_wmma_f32_16x16x4_f32(false, a, false, b,
                                                (short)0, acc, false, false);
  }
  for (int r = 0; r < 8; ++r) {
    int row = tm * 16 + ((lane < 16) ? r : (r + 8));
    Out[row * W_ + tn * 16 + n] = acc[r];
  }
}

// ---------------------------------------------------------------------------
// Main warp kernel: 1 thread per pixel; tap indices/weights computed once,
// then a 128-channel streaming gather with NT stores + prefetch.
// ---------------------------------------------------------------------------
__device__ __forceinline__ float upsample_half(const float* __restrict__ img,
                                               int yi, int xi) {
  float ys = (float)yi * (63.0f / 127.0f);
  float xs = (float)xi * (207.0f / 415.0f);
  int y0 = (int)floorf(ys); float wy = ys - (float)y0;
  int x0 = (int)floorf(xs); float wx = xs - (float)x0;
  int y1 = y0 + 1; if (y1 > HH_ - 1) y1 = HH_ - 1;
  int x1 = x0 + 1; if (x1 > WH_ - 1) x1 = WH_ - 1;
  float d00 = img[y0 * WH_ + x0], d01 = img[y0 * WH_ + x1];
  float d10 = img[y1 * WH_ + x0], d11 = img[y1 * WH_ + x1];
  float c0 = d00 * (1.0f - wy) + d10 * wy;
  float c1 = d01 * (1.0f - wy) + d11 * wy;
  return c0 * (1.0f - wx) + c1 * wx;
}

template <bool USE_WS>
__global__ void warp_kernel(const float* __restrict__ x,
                            const float* __restrict__ flow,
                            const float* __restrict__ dispH,
                            const float* __restrict__ k1,
                            const float* __restrict__ insz,
                            const float* __restrict__ expH,
                            const float* __restrict__ dispF,   // ws full-res
                            const float* __restrict__ expF,    // ws full-res
                            float* __restrict__ out) {
  int p = blockIdx.x * blockDim.x + threadIdx.x;
  if (p >= B_ * HW_) return;
  int b   = p / HW_;
  int rem = p - b * HW_;
  int yi  = rem / W_;
  int xi  = rem - yi * W_;

  // Intrinsics
  float sx = (float)W_ / insz[b * 2 + 1];
  float sy = (float)H_ / insz[b * 2 + 0];
  float fx = k1[b * 9 + 0] * sx;
  float fy = k1[b * 9 + 4] * sy;
  float cx = k1[b * 9 + 2] * sx;
  float cy = k1[b * 9 + 5] * sy;

  float d, e;
  if (USE_WS) {
    d = dispF[b * HW_ + rem];
    e = expF[b * HW_ + rem];
  } else {
    d = upsample_half(dispH + b * HWH_, yi, xi);
    e = upsample_half(expH + b * HWH_, yi, xi);
  }
  d *= (float)W_;                                   // disp * w_x

  float depth = fminf(fmaxf(fx * 0.54f / (d + 1e-8f), 1e-3f), 80.0f);
  float flx = flow[(b * 2 + 0) * HW_ + rem];
  float fly = flow[(b * 2 + 1) * HW_ + rem];

  float ptx = ((float)xi - cx) / fx * depth + flx;
  float pty = ((float)yi - cy) / fy * depth + fly;
  float ptz = depth + d * (e - 1.0f);

  float iz = 1.0f / (ptz + 1e-8f);
  float px = (fx * ptx + cx * ptz) * iz;
  float py = (fy * pty + cy * ptz) * iz;

  // grid_sample coords (align_corners=False, zeros padding)
  float gx = px * ((float)W_ / (float)(W_ - 1)) - 0.5f;
  float gy = py * ((float)H_ / (float)(H_ - 1)) - 0.5f;

  float x0f = floorf(gx), y0f = floorf(gy);
  float wx = gx - x0f,    wy = gy - y0f;
  float x1f = x0f + 1.0f, y1f = y0f + 1.0f;

  bool vx0 = (x0f >= 0.0f) && (x0f <= (float)(W_ - 1));
  bool vx1 = (x1f >= 0.0f) && (x1f <= (float)(W_ - 1));
  bool vy0 = (y0f >= 0.0f) && (y0f <= (float)(H_ - 1));
  bool vy1 = (y1f >= 0.0f) && (y1f <= (float)(H_ - 1));

  int ix0 = (int)fminf(fmaxf(x0f, 0.0f), (float)(W_ - 1));
  int ix1 = (int)fminf(fmaxf(x1f, 0.0f), (float)(W_ - 1));
  int iy0 = (int)fminf(fmaxf(y0f, 0.0f), (float)(H_ - 1));
  int iy1 = (int)fminf(fmaxf(y1f, 0.0f), (float)(H_ - 1));

  float w00 = (1.0f - wx) * (1.0f - wy) * ((vx0 && vy0) ? 1.0f : 0.0f);
  float w10 = wx * (1.0f - wy)          * ((vx1 && vy0) ? 1.0f : 0.0f);
  float w01 = (1.0f - wx) * wy          * ((vx0 && vy1) ? 1.0f : 0.0f);
  float w11 = wx * wy                   * ((vx1 && vy1) ? 1.0f : 0.0f);

  float maskv = w00 + w10 + w01 + w11;              // grid_sample of ones
  float keep  = (maskv >= 1.0f) ? 1.0f : 0.0f;
  w00 *= keep; w10 *= keep; w01 *= keep; w11 *= keep;

  int i00 = iy0 * W_ + ix0;
  int i10 = iy0 * W_ + ix1;
  int i01 = iy1 * W_ + ix0;
  int i11 = iy1 * W_ + ix1;

  const float* xb = x + (size_t)b * C_ * HW_;
  float*       ob = out + (size_t)b * C_ * HW_ + rem;

#pragma unroll 4
  for (int c = 0; c < C_; ++c) {
    const float* xp = xb + c * HW_;
    if ((c & 7) == 0 && (c + 8) < C_) {             // stage taps 8 planes ahead
      const float* xq = xp + 8 * HW_;
      __builtin_prefetch(xq + i00, 0, 1);
      __builtin_prefetch(xq + i11, 0, 1);
    }
    float v = w00 * xp[i00] + w10 * xp[i10] + w01 * xp[i01] + w11 * xp[i11];
    __builtin_nontemporal_store(v, ob + c * HW_);   // write-once output
  }
}

extern "C" void kernel_launch(void* const* d_in, const int* in_sizes, int n_in,
                              void* d_out, int out_size, void* d_ws, size_t ws_size,
                              hipStream_t stream) {
  const float* x    = (const float*)d_in[0];
  const float* flow = (const float*)d_in[1];
  const float* disp = (const float*)d_in[2];
  const float* k1   = (const float*)d_in[3];
  const float* insz = (const float*)d_in[4];
  const float* expm = (const float*)d_in[5];
  float* out = (float*)d_out;
  float* ws  = (float*)d_ws;

  const int npix = B_ * HW_;                        // 212992 = 832 * 256
  const bool use_ws = (ws_size >= (size_t)WS_FLOATS * sizeof(float));

  if (use_ws) {
    // Zero Ry + RxT, then build interpolation matrices.
    hipMemsetAsync(ws, 0, (size_t)(RXT_OFF + WH_ * W_) * sizeof(float), stream);
    build_interp<<<3, 256, 0, stream>>>(ws);
    // Upsample disp/exp as two batched WMMA GEMMs.
    resize_gemm1<<<dim3(104, 8), dim3(32), 0, stream>>>(disp, expm, ws);
    resize_gemm2<<<dim3(208, 8), dim3(32), 0, stream>>>(ws);
    warp_kernel<true><<<npix / 256, 256, 0, stream>>>(
        x, flow, disp, k1, insz, expm,
        ws + FULL_OFF, ws + FULL_OFF + 4 * HW_, out);
  } else {
    warp_kernel<false><<<npix / 256, 256, 0, stream>>>(
        x, flow, disp, k1, insz, expm, nullptr, nullptr, out);
  }
}